// MinkUNetBase_63496796504727
// MI455X (gfx1250) — compile-verified
//
#include <hip/hip_runtime.h>
#include <hip/hip_bf16.h>

// ---------------------------------------------------------------------------
// MinkUNet (dense 48^3 grid, fp32) on gfx1250.
// All convolutions are implicit GEMM on V_WMMA_F32_16X16X4_F32 (wave32).
// Per wave: M=32 out-channels x N=32 voxels via 4 accumulators (4 WMMAs per
// K-step fed by 2x b64 weight loads + 4x b32 input loads, no div/mod, no
// 64-bit muls in the hot loop — only pointer bumps).
// K is ordered tap-major: boundary check + base address once per kernel tap,
// inner loop over Cin (always a multiple of 4: conv0's 3-channel input is
// zero-padded to 4 channels by a tiny pad kernel, eliminating the slow path).
// Weights are pre-repacked to [tap][Co][CinP] (zero padded) so A is a single
// 64-bit load and padded K-slots self-nullify. Loads are unconditional from
// clamped addresses; only the spatial-OOB zero on B needs v_cndmask (WMMA
// rows/cols are independent, so garbage in never-stored rows/cols is harmless).
// Tiles are co-major so the 8 waves of a block share B addresses (cache reuse).
// BatchNorm (training stats) = LDS-tree reduction + fused apply(+res,+relu).
// Channel concat = contiguous regions (NCDHW, N=1) via async d2d copies.
// ---------------------------------------------------------------------------

typedef __attribute__((ext_vector_type(2))) float v2f;
typedef __attribute__((ext_vector_type(8))) float v8f;

#define WMMA_F32(A, B, C) \
  __builtin_amdgcn_wmma_f32_16x16x4_f32(false, (A), false, (B), (short)0, (C), false, false)

// ---------------- input pad (conv0 only: 3ch -> 4ch, zero fill) ----------------

__global__ __launch_bounds__(256) void pad_input(
    const float* __restrict__ x, float* __restrict__ xp, int Scopy, int Stotal)
{
  int i = blockIdx.x * 256 + threadIdx.x;
  if (i < Stotal) xp[i] = (i < Scopy) ? x[i] : 0.0f;
}

// ---------------- weight repack ----------------

// conv weights (Co, Cin, k,k,k) -> W2[((tap*Co + co)*CinP) + ci], zero padded.
__global__ __launch_bounds__(256) void repack_w_conv(
    const float* __restrict__ Wt, float* __restrict__ W2,
    int Co, int Cin, int CinP, int kvol, int total)
{
  int i = blockIdx.x * 256 + threadIdx.x;
  if (i >= total) return;
  int tap = i / (Co * CinP);
  int r   = i - tap * (Co * CinP);
  int co  = r / CinP;
  int ci  = r - co * CinP;
  W2[i] = (ci < Cin) ? Wt[((size_t)co * Cin + ci) * kvol + tap] : 0.0f;
}

// convT weights (Cin, Co, 2,2,2) -> W2[((tap*Co + co)*Cin) + ci] (Cin % 4 == 0).
__global__ __launch_bounds__(256) void repack_w_convT(
    const float* __restrict__ Wt, float* __restrict__ W2, int Co, int Cin, int total)
{
  int i = blockIdx.x * 256 + threadIdx.x;
  if (i >= total) return;
  int tap = i / (Co * Cin);
  int r   = i - tap * (Co * Cin);
  int co  = r / Cin;
  int ci  = r - co * Cin;
  W2[i] = Wt[((size_t)ci * Co + co) * 8 + tap];
}

// ---------------- implicit-GEMM conv (M=32 x N=32 per wave) ----------------

__global__ __launch_bounds__(256) void conv3d_wmma_f32(
    const float* __restrict__ X, const float* __restrict__ W2,
    const float* __restrict__ bias, float* __restrict__ Y,
    int Cin, int Co, int D, int H, int W,
    int Do, int Ho, int Wo, int ksz, int strd, int pad,
    int nCoT, int nTiles)
{
  int wave = blockIdx.x * 8 + ((int)threadIdx.x >> 5);
  if (wave >= nTiles) return;               // wave-uniform: EXEC stays all-1s
  int coT  = wave % nCoT;                   // co-major: block waves share voxels
  int voxT = wave / nCoT;

  int lane = (int)threadIdx.x & 31;
  int m = lane & 15;
  int g = lane >> 4;

  int So = Do * Ho * Wo;
  int HoWo = Ho * Wo;
  int n0 = voxT * 32 + m;                   // two column streams
  int n1 = n0 + 16;
  bool vn0 = n0 < So, vn1 = n1 < So;
  int nc0 = vn0 ? n0 : 0;                   // garbage columns never stored
  int nc1 = vn1 ? n1 : 0;
  int od0 = nc0 / HoWo; int r0 = nc0 - od0 * HoWo; int oh0 = r0 / Wo; int ow0 = r0 - oh0 * Wo;
  int od1 = nc1 / HoWo; int r1 = nc1 - od1 * HoWo; int oh1 = r1 / Wo; int ow1 = r1 - oh1 * Wo;
  int idb0 = od0 * strd - pad, ihb0 = oh0 * strd - pad, iwb0 = ow0 * strd - pad;
  int idb1 = od1 * strd - pad, ihb1 = oh1 * strd - pad, iwb1 = ow1 * strd - pad;

  int coA0 = coT * 32 + m;
  int coA1 = coA0 + 16;
  int cA0 = coA0 < Co ? coA0 : Co - 1;      // clamp: garbage rows never stored
  int cA1 = coA1 < Co ? coA1 : Co - 1;

  size_t S = (size_t)D * H * W;             // per-channel spatial size
  size_t S4 = 4 * S;

  v8f acc00 = {}, acc01 = {}, acc10 = {}, acc11 = {};

  int tap = 0;
  for (int kd = 0; kd < ksz; ++kd) {
    int id0 = idb0 + kd, id1 = idb1 + kd;
    for (int kh = 0; kh < ksz; ++kh) {
      int ih0 = ihb0 + kh, ih1 = ihb1 + kh;
      for (int kw = 0; kw < ksz; ++kw, ++tap) {
        int iw0 = iwb0 + kw, iw1 = iwb1 + kw;
        bool ok0 = ((unsigned)id0 < (unsigned)D) & ((unsigned)ih0 < (unsigned)H) &
                   ((unsigned)iw0 < (unsigned)W);
        bool ok1 = ((unsigned)id1 < (unsigned)D) & ((unsigned)ih1 < (unsigned)H) &
                   ((unsigned)iw1 < (unsigned)W);
        int base0 = ok0 ? ((id0 * H + ih0) * W + iw0) : 0;
        int base1 = ok1 ? ((id1 * H + ih1) * W + iw1) : 0;

        const float* w0 = W2 + ((size_t)tap * Co + cA0) * Cin + 2 * g;
        const float* w1 = W2 + ((size_t)tap * Co + cA1) * Cin + 2 * g;
        const float* p0 = X + base0 + (size_t)(2 * g) * S;  // k-slot 2g, col n0
        const float* q0 = p0 + S;                           // k-slot 2g+1
        const float* p1 = X + base1 + (size_t)(2 * g) * S;  // col n1
        const float* q1 = p1 + S;

        for (int ci = 0; ci < Cin; ci += 4) {
          v2f a0 = *(const v2f*)w0;
          v2f a1 = *(const v2f*)w1;
          float b0x = *p0, b0y = *q0, b1x = *p1, b1y = *q1;
          v2f b0, b1;
          b0.x = ok0 ? b0x : 0.0f;          // spatial-OOB zero (per column)
          b0.y = ok0 ? b0y : 0.0f;
          b1.x = ok1 ? b1x : 0.0f;
          b1.y = ok1 ? b1y : 0.0f;
          acc00 = WMMA_F32(a0, b0, acc00);
          acc10 = WMMA_F32(a1, b0, acc10);
          acc01 = WMMA_F32(a0, b1, acc01);
          acc11 = WMMA_F32(a1, b1, acc11);
          w0 += 4; w1 += 4;
          p0 += S4; q0 += S4; p1 += S4; q1 += S4;
        }
      }
    }
  }

#pragma unroll
  for (int rr = 0; rr < 8; ++rr) {
    int co  = coT * 32 + rr + 8 * g;
    int co1 = co + 16;
    float bo0 = bias ? bias[co  < Co ? co  : 0] : 0.0f;
    float bo1 = bias ? bias[co1 < Co ? co1 : 0] : 0.0f;
    if (co < Co) {
      if (vn0) Y[(size_t)co * So + n0] = acc00[rr] + bo0;
      if (vn1) Y[(size_t)co * So + n1] = acc01[rr] + bo0;
    }
    if (co1 < Co) {
      if (vn0) Y[(size_t)co1 * So + n0] = acc10[rr] + bo1;
      if (vn1) Y[(size_t)co1 * So + n1] = acc11[rr] + bo1;
    }
  }
}

// ---------------- transposed conv k2 s2 (M=32 x N=32 per wave) ----------------
// out[co][2d+a][2h+b][2w+c] = sum_ci x[ci][d][h][w] * Wt[ci][co][a][b][c].
// Waves tiled by (coTile, tap, input-voxel tile): tap is wave-uniform, B loads
// fully coalesced, no selects at all in the inner loop (Cin % 4 == 0 always).
__global__ __launch_bounds__(256) void convT_k2s2_wmma_f32(
    const float* __restrict__ X, const float* __restrict__ W2,
    float* __restrict__ Y,
    int Cin, int Co, int D, int H, int W, int nCoT, int nTiles)
{
  int wave = blockIdx.x * 8 + ((int)threadIdx.x >> 5);
  if (wave >= nTiles) return;
  int coT  = wave % nCoT;
  int rem  = wave / nCoT;
  int tap  = rem & 7;
  int voxT = rem >> 3;
  int ta = tap >> 2, tb = (tap >> 1) & 1, tc = tap & 1;

  int lane = (int)threadIdx.x & 31;
  int m = lane & 15, g = lane >> 4;

  int Sin = D * H * W;
  int n0 = voxT * 32 + m;                   // input voxels (two column streams)
  int n1 = n0 + 16;
  bool vn0 = n0 < Sin, vn1 = n1 < Sin;
  int nc0 = vn0 ? n0 : 0;
  int nc1 = vn1 ? n1 : 0;

  int coA0 = coT * 32 + m;
  int coA1 = coA0 + 16;
  int cA0 = coA0 < Co ? coA0 : Co - 1;
  int cA1 = coA1 < Co ? coA1 : Co - 1;

  size_t Ss = (size_t)Sin, S4 = 4 * Ss;
  const float* w0 = W2 + ((size_t)tap * Co + cA0) * Cin + 2 * g;
  const float* w1 = W2 + ((size_t)tap * Co + cA1) * Cin + 2 * g;
  const float* p0 = X + nc0 + (size_t)(2 * g) * Ss;
  const float* q0 = p0 + Ss;
  const float* p1 = X + nc1 + (size_t)(2 * g) * Ss;
  const float* q1 = p1 + Ss;

  v8f acc00 = {}, acc01 = {}, acc10 = {}, acc11 = {};
  for (int ci = 0; ci < Cin; ci += 4) {
    v2f a0 = *(const v2f*)w0;
    v2f a1 = *(const v2f*)w1;
    v2f b0, b1;
    b0.x = *p0; b0.y = *q0;
    b1.x = *p1; b1.y = *q1;
    acc00 = WMMA_F32(a0, b0, acc00);
    acc10 = WMMA_F32(a1, b0, acc10);
    acc01 = WMMA_F32(a0, b1, acc01);
    acc11 = WMMA_F32(a1, b1, acc11);
    w0 += 4; w1 += 4;
    p0 += S4; q0 += S4; p1 += S4; q1 += S4;
  }

  int HW = H * W;
  int d0 = nc0 / HW; int r0 = nc0 - d0 * HW; int h0 = r0 / W; int w0i = r0 - h0 * W;
  int d1 = nc1 / HW; int r1 = nc1 - d1 * HW; int h1 = r1 / W; int w1i = r1 - h1 * W;
  int Ho = 2 * H, Wo = 2 * W;
  size_t o0 = ((size_t)(2 * d0 + ta) * Ho + (2 * h0 + tb)) * Wo + (2 * w0i + tc);
  size_t o1 = ((size_t)(2 * d1 + ta) * Ho + (2 * h1 + tb)) * Wo + (2 * w1i + tc);
  size_t Sout = 8 * Ss;
#pragma unroll
  for (int rr = 0; rr < 8; ++rr) {
    int co  = coT * 32 + rr + 8 * g;
    int co1 = co + 16;
    if (co < Co) {
      if (vn0) Y[(size_t)co * Sout + o0] = acc00[rr];
      if (vn1) Y[(size_t)co * Sout + o1] = acc01[rr];
    }
    if (co1 < Co) {
      if (vn0) Y[(size_t)co1 * Sout + o0] = acc10[rr];
      if (vn1) Y[(size_t)co1 * Sout + o1] = acc11[rr];
    }
  }
}

// ---------------- batch norm ----------------

// Training-mode BN statistics: one workgroup per channel, LDS tree reduction.
__global__ __launch_bounds__(256) void bn_stats(
    const float* __restrict__ X, float* __restrict__ stats, int S)
{
  int c = blockIdx.x;
  const float* p = X + (size_t)c * S;
  float s = 0.0f, s2 = 0.0f;
  for (int i = threadIdx.x; i < S; i += 256) {
    float v = p[i];
    s += v; s2 += v * v;
  }
  __shared__ float sh[512];
  sh[threadIdx.x]       = s;
  sh[256 + threadIdx.x] = s2;
  __syncthreads();
  for (int off = 128; off > 0; off >>= 1) {
    if ((int)threadIdx.x < off) {
      sh[threadIdx.x]       += sh[threadIdx.x + off];
      sh[256 + threadIdx.x] += sh[256 + threadIdx.x + off];
    }
    __syncthreads();
  }
  if (threadIdx.x == 0) {
    float inv = 1.0f / (float)S;
    float mean = sh[0] * inv;
    float var  = sh[256] * inv - mean * mean;
    if (var < 0.0f) var = 0.0f;
    stats[2 * c]     = mean;
    stats[2 * c + 1] = rsqrtf(var + 1e-5f);
  }
}

// In-place BN apply fused with optional residual add and ReLU.
// Channel = blockIdx.y (no per-element division).
__global__ __launch_bounds__(256) void bn_apply(
    float* __restrict__ X, const float* __restrict__ res,
    const float* __restrict__ gam, const float* __restrict__ bet,
    const float* __restrict__ stats, int S, int relu)
{
  int i = blockIdx.x * 256 + threadIdx.x;
  if (i >= S) return;
  int c = blockIdx.y;
  size_t idx = (size_t)c * S + i;
  float v = (X[idx] - stats[2 * c]) * stats[2 * c + 1] * gam[c] + bet[c];
  if (res) v += res[idx];
  if (relu) v = fmaxf(v, 0.0f);
  X[idx] = v;
}

// ---------------- host orchestration ----------------

static inline int cdiv(int a, int b) { return (a + b - 1) / b; }

extern "C" void kernel_launch(void* const* d_in, const int* in_sizes, int n_in,
                              void* d_out, int out_size, void* d_ws, size_t ws_size,
                              hipStream_t stream)
{
  (void)in_sizes; (void)n_in; (void)out_size; (void)ws_size;

  // d_in flattening assumed to follow jax pytree order: dict keys sorted
  // alphabetically, lists in order, bn dicts {'b','g'} -> b then g.
  // Leaf index map (x=0, then 146 parameter leaves):
  //   block1 bb0=1 bb1=7 | block2 13(ds),22 | block3 28(ds),37 | block4 43(ds),52
  //   block5 58(ds),67 | block6 73(ds),82 | block7 88(ds),97 | block8 103(ds),112
  //   bn0=118 bn1=120 bn2=122 bn3=124 bn4=126 | bntr4=128..bntr7=134
  //   conv0=136..conv4=140 | convtr4=141..convtr7=144 | final_b=145 final_w=146
  // Each bb: [bn1.b, bn1.g, bn2.b, bn2.g, conv1, conv2, (ds_bn.b, ds_bn.g, ds_conv)]
  auto F = [&](int i) { return (const float*)d_in[i]; };
  const float* x0 = (const float*)d_in[0];

  const int S1 = 48 * 48 * 48;   // 110592
  const int S2 = 24 * 24 * 24;   // 13824
  const int S4 = 12 * 12 * 12;   // 1728
  const int S8 = 6 * 6 * 6;      // 216

  // Workspace arena (floats).
  float* ws = (float*)d_ws;
  size_t off = 0;
  float* stats = ws + off; off += 1024;               // 2 * maxC(=256) w/ slack
  float* WREP  = ws + off; off += 2700000;            // repacked weights (max 256*384*27)
  float* XPAD  = ws + off; off += (size_t)4 * S1;     // conv0 input padded to 4ch
  float* Pp1 = ws + off; off += (size_t)32  * S1;     // skip out_p1
  float* Pb1 = ws + off; off += (size_t)32  * S2;     // skip b1p2
  float* Pb2 = ws + off; off += (size_t)64  * S4;     // skip b2p4
  float* Pb3 = ws + off; off += (size_t)128 * S8;     // skip b3p8
  float* BIG = ws + off; off += (size_t)128 * S1;     // concat / stage inputs
  float* TA  = ws + off; off += (size_t)96  * S1;     // rotating temporaries
  float* TB  = ws + off; off += (size_t)96  * S1;
  float* TC  = ws + off; off += (size_t)96  * S1;

  // X must have CinP (=Cin rounded to x4) channels; conv0 uses the padded copy.
  auto run_conv = [&](const float* X, const float* Wt, const float* bias, float* Y,
                      int Cin, int Co, int D, int ksz, int strd, int pad, int Do) {
    int kvol = ksz * ksz * ksz;
    int CinP = (Cin + 3) & ~3;
    int wtot = kvol * Co * CinP;
    repack_w_conv<<<cdiv(wtot, 256), 256, 0, stream>>>(Wt, WREP, Co, Cin, CinP, kvol, wtot);
    int So = Do * Do * Do;
    int nCoT = cdiv(Co, 32);
    int nTiles = nCoT * cdiv(So, 32);
    conv3d_wmma_f32<<<cdiv(nTiles, 8), 256, 0, stream>>>(
        X, WREP, bias, Y, CinP, Co, D, D, D, Do, Do, Do, ksz, strd, pad, nCoT, nTiles);
  };

  auto run_bn = [&](float* Xio, const float* res, const float* gam, const float* bet,
                    int C, int S, int relu) {
    bn_stats<<<C, 256, 0, stream>>>(Xio, stats, S);
    dim3 grid(cdiv(S, 256), C);
    bn_apply<<<grid, 256, 0, stream>>>(Xio, res, gam, bet, stats, S, relu);
  };

  // conv -> bn-stats -> fused bn-apply (in place)
  auto convbn = [&](const float* X, float* Yt, int wi, int gi, int bi,
                    int Cin, int Co, int D, int ksz, int strd,
                    const float* res, int relu) {
    int pad = (strd == 1) ? (ksz - 1) / 2 : 0;        // SAME for s1, VALID for s2
    int Do  = (strd == 1) ? D : D / 2;
    run_conv(X, F(wi), nullptr, Yt, Cin, Co, D, ksz, strd, pad, Do);
    run_bn(Yt, res, F(gi), F(bi), Co, Do * Do * Do, relu);
  };

  auto convTbn = [&](const float* X, float* Yt, int wi, int gi, int bi,
                     int Cin, int Co, int D) {
    int wtot = 8 * Co * Cin;
    repack_w_convT<<<cdiv(wtot, 256), 256, 0, stream>>>(F(wi), WREP, Co, Cin, wtot);
    int Sin = D * D * D;
    int nCoT = cdiv(Co, 32);
    int nTiles = nCoT * 8 * cdiv(Sin, 32);
    convT_k2s2_wmma_f32<<<cdiv(nTiles, 8), 256, 0, stream>>>(
        X, WREP, Yt, Cin, Co, D, D, D, nCoT, nTiles);
    run_bn(Yt, nullptr, F(gi), F(bi), Co, 8 * Sin, 1);
  };

  // BasicBlock: relu(bn(conv3(x))) -> bn(conv3(.)) + residual -> relu.
  // Output lands in t2; t3 only used for the 1x1 downsample path.
  auto bblock = [&](const float* xin, float* t1, float* t2, float* t3,
                    int base, bool ds, int Cin, int Co, int D) {
    convbn(xin, t1, base + 4, base + 1, base + 0, Cin, Co, D, 3, 1, nullptr, 1);
    const float* res = xin;
    if (ds) {
      convbn(xin, t3, base + 8, base + 7, base + 6, Cin, Co, D, 1, 1, nullptr, 0);
      res = t3;
    }
    convbn(t1, t2, base + 5, base + 3, base + 2, Co, Co, D, 3, 1, res, 1);
  };

  auto d2d = [&](float* dst, const float* src, size_t nfloat) {
    hipMemcpyAsync(dst, src, nfloat * sizeof(float), hipMemcpyDeviceToDevice, stream);
  };

  // -------------------- encoder --------------------
  pad_input<<<cdiv(4 * S1, 256), 256, 0, stream>>>(x0, XPAD, 3 * S1, 4 * S1);
  convbn(XPAD, Pp1, 136, 119, 118, 3, 32, 48, 5, 1, nullptr, 1);        // out_p1 (32,48^3)
  convbn(Pp1, TA, 137, 121, 120, 32, 32, 48, 2, 2, nullptr, 1);         // (32,24^3)
  bblock(TA, TB, TC, nullptr, 1, false, 32, 32, 24);                    // -> TC
  bblock(TC, TA, TB, nullptr, 7, false, 32, 32, 24);                    // -> TB = b1p2
  d2d(Pb1, TB, (size_t)32 * S2);

  convbn(TB, TA, 138, 123, 122, 32, 32, 24, 2, 2, nullptr, 1);          // (32,12^3)
  bblock(TA, TB, TC, BIG, 13, true, 32, 64, 12);                        // -> TC
  bblock(TC, TA, TB, nullptr, 22, false, 64, 64, 12);                   // -> TB = b2p4
  d2d(Pb2, TB, (size_t)64 * S4);

  convbn(TB, TA, 139, 125, 124, 64, 64, 12, 2, 2, nullptr, 1);          // (64,6^3)
  bblock(TA, TB, TC, BIG, 28, true, 64, 128, 6);                        // -> TC
  bblock(TC, TA, TB, nullptr, 37, false, 128, 128, 6);                  // -> TB = b3p8
  d2d(Pb3, TB, (size_t)128 * S8);

  convbn(TB, TA, 140, 127, 126, 128, 128, 6, 2, 2, nullptr, 1);         // (128,3^3)
  bblock(TA, TB, TC, BIG, 43, true, 128, 256, 3);                       // -> TC
  bblock(TC, TA, TB, nullptr, 52, false, 256, 256, 3);                  // -> TB (256,3^3)

  // -------------------- decoder --------------------
  convTbn(TB, BIG, 141, 129, 128, 256, 256, 3);                         // (256,6^3) in BIG
  d2d(BIG + (size_t)256 * S8, Pb3, (size_t)128 * S8);                   // concat -> (384,6^3)
  bblock(BIG, TA, TB, TC, 58, true, 384, 256, 6);                       // -> TB
  bblock(TB, TA, TC, nullptr, 67, false, 256, 256, 6);                  // -> TC

  convTbn(TC, BIG, 142, 131, 130, 256, 128, 6);                         // (128,12^3)
  d2d(BIG + (size_t)128 * S4, Pb2, (size_t)64 * S4);                    // concat -> (192,12^3)
  bblock(BIG, TA, TB, TC, 73, true, 192, 128, 12);                      // -> TB
  bblock(TB, TA, TC, nullptr, 82, false, 128, 128, 12);                 // -> TC

  convTbn(TC, BIG, 143, 133, 132, 128, 96, 12);                         // (96,24^3)
  d2d(BIG + (size_t)96 * S2, Pb1, (size_t)32 * S2);                     // concat -> (128,24^3)
  bblock(BIG, TA, TB, TC, 88, true, 128, 96, 24);                       // -> TB
  bblock(TB, TA, TC, nullptr, 97, false, 96, 96, 24);                   // -> TC

  convTbn(TC, BIG, 144, 135, 134, 96, 96, 24);                          // (96,48^3)
  d2d(BIG + (size_t)96 * S1, Pp1, (size_t)32 * S1);                     // concat -> (128,48^3)
  bblock(BIG, TA, TB, TC, 103, true, 128, 96, 48);                      // -> TB
  bblock(TB, TA, TC, nullptr, 112, false, 96, 96, 48);                  // -> TC

  // final 1x1 conv + bias -> d_out (20,48^3)
  run_conv(TC, F(146), F(145), (float*)d_out, 96, 20, 48, 1, 1, 0, 48);
}